// PKM_36000415875513
// MI455X (gfx1250) — compile-verified
//
#include <hip/hip_runtime.h>

// ---------------- problem constants ----------------
#define DIM_IN   1024
#define DIM_QF   2048      // DIM_HIDDEN * NUM_HEADS
#define NSUB     256
#define TOPK     32
#define NHEAD    4
#define NTOK     1024      // 4 * 256 tokens
#define SUBKEY   256       // DIM_HIDDEN / 2
#define BN_EPS   1e-5f
#define NEG_BIG  (-3.402823466e38f)
#define ROWCHUNK 16        // BN partial-reduction row chunks (1024/64)

typedef __attribute__((ext_vector_type(2))) float v2f;
typedef __attribute__((ext_vector_type(8))) float v8f;

// ---------------------------------------------------------------------------
// fp32 WMMA, one wave computes a 16x32 tile: D0|D1 = A[16xK] * (B0|B1)[32xK]^T
// Two independent accumulator chains (ILP) + A-fragment reuse across both.
// V_WMMA_F32_16X16X4_F32 layouts (ISA 7.12.2):
//   A: lanes 0-15 -> row=lane, K=0,1 ; lanes 16-31 -> row=lane-16, K=2,3
//   B operand Bw[k][n] = Bmem[n][k] -> lane loads Bmem[l16][k + half*2 .. +1]
//   D: VGPR r, lanes 0-15 -> row r ; lanes 16-31 -> row r+8
// ---------------------------------------------------------------------------
__device__ __forceinline__ void wmma_f32_tile2(const float* __restrict__ A, int lda,
                                               const float* __restrict__ B, int ldb,
                                               int K, v8f& acc0, v8f& acc1) {
    const int lane = threadIdx.x & 31;
    const int half = lane >> 4;
    const int l16  = lane & 15;
    const float* ap  = A + l16 * lda + half * 2;
    const float* bp0 = B + l16 * ldb + half * 2;
    const float* bp1 = B + (16 + l16) * ldb + half * 2;
    for (int k = 0; k < K; k += 4) {
        v2f a  = *(const v2f*)(ap  + k);
        v2f b0 = *(const v2f*)(bp0 + k);
        v2f b1 = *(const v2f*)(bp1 + k);
        acc0 = __builtin_amdgcn_wmma_f32_16x16x4_f32(false, a, false, b0,
                                                     (short)0, acc0, false, false);
        acc1 = __builtin_amdgcn_wmma_f32_16x16x4_f32(false, a, false, b1,
                                                     (short)0, acc1, false, false);
    }
}

// ---------------- kernel 1: q = x @ Wq^T  (1024x1024 * 1024x2048) ----------
__global__ __launch_bounds__(32)
void PKM_qproj(const float* __restrict__ x, const float* __restrict__ Wq,
               float* __restrict__ q) {
    const int m0 = blockIdx.x * 16;     // token tile
    const int n0 = blockIdx.y * 32;     // output-feature tile (16x32 per wave)
    v8f acc0 = {}, acc1 = {};
    wmma_f32_tile2(x + m0 * DIM_IN, DIM_IN, Wq + n0 * DIM_IN, DIM_IN, DIM_IN,
                   acc0, acc1);
    const int lane = threadIdx.x & 31, half = lane >> 4, l16 = lane & 15;
#pragma unroll
    for (int r = 0; r < 8; ++r) {
        const int row = (m0 + r + half * 8) * DIM_QF;
        q[row + n0 + l16]      = acc0[r];
        q[row + n0 + 16 + l16] = acc1[r];
    }
}

// ---------------- kernel 2a: BN partial sums (16 row-chunks) ---------------
__global__ __launch_bounds__(256)
void PKM_bn_partial(const float* __restrict__ q,
                    float* __restrict__ psum, float* __restrict__ psum2) {
    const int col   = blockIdx.x * blockDim.x + threadIdx.x;  // 2048 columns
    const int chunk = blockIdx.y;                             // 16 chunks of 64 rows
    float s = 0.f, s2 = 0.f;
    const int r0 = chunk * (NTOK / ROWCHUNK);
    for (int r = 0; r < NTOK / ROWCHUNK; ++r) {
        float v = q[(r0 + r) * DIM_QF + col];
        s += v; s2 += v * v;
    }
    psum [chunk * DIM_QF + col] = s;
    psum2[chunk * DIM_QF + col] = s2;
}

// ---------------- kernel 2b: finalize BN stats -> scale/shift --------------
__global__ __launch_bounds__(256)
void PKM_bn_final(const float* __restrict__ psum, const float* __restrict__ psum2,
                  const float* __restrict__ gamma, const float* __restrict__ beta,
                  float* __restrict__ scale, float* __restrict__ shift) {
    const int col = blockIdx.x * blockDim.x + threadIdx.x;    // 2048 columns
    float s = 0.f, s2 = 0.f;
#pragma unroll
    for (int c = 0; c < ROWCHUNK; ++c) {
        s  += psum [c * DIM_QF + col];
        s2 += psum2[c * DIM_QF + col];
    }
    const float mean = s * (1.0f / NTOK);
    const float var  = s2 * (1.0f / NTOK) - mean * mean;
    const float sc   = rsqrtf(var + BN_EPS) * gamma[col];
    scale[col] = sc;
    shift[col] = beta[col] - mean * sc;
}

// ---------------- kernel 3: apply BN in-place (float4) ---------------------
__global__ __launch_bounds__(256)
void PKM_bn_apply(float* __restrict__ q,
                  const float* __restrict__ scale, const float* __restrict__ shift) {
    const int i4 = blockIdx.x * blockDim.x + threadIdx.x;    // 524288 float4
    const int c4 = i4 & (DIM_QF / 4 - 1);
    float4 v  = ((const float4*)q)[i4];
    float4 sc = ((const float4*)scale)[c4];
    float4 sh = ((const float4*)shift)[c4];
    v.x = v.x * sc.x + sh.x;  v.y = v.y * sc.y + sh.y;
    v.z = v.z * sc.z + sh.z;  v.w = v.w * sc.w + sh.w;
    ((float4*)q)[i4] = v;
}

// ---------------- kernel 4: sub-key scores (WMMA, per head/side) -----------
__global__ __launch_bounds__(32)
void PKM_scores(const float* __restrict__ qn,
                const float* __restrict__ keyl, const float* __restrict__ keyr,
                float* __restrict__ scores) {
    const int m0 = blockIdx.x * 16;        // token tile
    const int n0 = blockIdx.y * 32;        // subkey tile (within 256)
    const int z  = blockIdx.z;             // h*2 + side
    const int h  = z >> 1;
    const float* key = ((z & 1) ? keyr : keyl) + h * NSUB * SUBKEY;
    v8f acc0 = {}, acc1 = {};
    wmma_f32_tile2(qn + m0 * DIM_QF + z * SUBKEY, DIM_QF,
                   key + n0 * SUBKEY, SUBKEY, SUBKEY, acc0, acc1);
    const int lane = threadIdx.x & 31, half = lane >> 4, l16 = lane & 15;
#pragma unroll
    for (int r = 0; r < 8; ++r) {
        const int row = (m0 + r + half * 8) * DIM_QF + z * NSUB;
        scores[row + n0 + l16]      = acc0[r];
        scores[row + n0 + 16 + l16] = acc1[r];
    }
}

// ---------------- kernel 5: top-k, pair top-k, softmax, gather -------------
__global__ __launch_bounds__(256)
void PKM_topk_gather(const float* __restrict__ scores,
                     const float* __restrict__ vtab,
                     float* __restrict__ out) {
    __shared__ float s_topv[NHEAD * 2][TOPK];
    __shared__ int   s_topi[NHEAD * 2][TOPK];
    __shared__ float s_w [NHEAD][TOPK];
    __shared__ int   s_vi[NHEAD][TOPK];

    const int token = blockIdx.x;
    const int tid   = threadIdx.x;
    const int wave  = tid >> 5;
    const int lane  = tid & 31;

    // ---- stage 1: each wave does top-32-of-256 for one (head, side) ----
    {
        const float* sc = scores + token * DIM_QF + wave * NSUB;
        const int base = lane * 8;
        float v[8];
        {   // two coalesced float4 loads per lane
            float4 a = ((const float4*)(sc + base))[0];
            float4 b = ((const float4*)(sc + base))[1];
            v[0]=a.x; v[1]=a.y; v[2]=a.z; v[3]=a.w;
            v[4]=b.x; v[5]=b.y; v[6]=b.z; v[7]=b.w;
        }
        for (int it = 0; it < TOPK; ++it) {
            float bv = v[0]; int bj = 0;
#pragma unroll
            for (int j = 1; j < 8; ++j)
                if (v[j] > bv) { bv = v[j]; bj = j; }
            int bi = base + bj;
#pragma unroll
            for (int off = 16; off > 0; off >>= 1) {   // wave32 argmax, ties -> lower idx
                float ov = __shfl_xor(bv, off, 32);
                int   oi = __shfl_xor(bi, off, 32);
                if (ov > bv || (ov == bv && oi < bi)) { bv = ov; bi = oi; }
            }
            if ((bi >> 3) == lane) {                   // owner removes the winner
                const int jw = bi & 7;
#pragma unroll
                for (int j = 0; j < 8; ++j)
                    if (j == jw) v[j] = NEG_BIG;
            }
            if (lane == 0) { s_topv[wave][it] = bv; s_topi[wave][it] = bi; }
        }
    }
    __syncthreads();

    // ---- stage 2: waves 0..3 -> top-32 of 32x32 pair sums + softmax ----
    if (wave < NHEAD) {
        const int h = wave;
        const float vl = s_topv[h * 2][lane];          // my left score (row = lane)
        float tr[TOPK];
#pragma unroll
        for (int j = 0; j < TOPK; ++j) tr[j] = s_topv[h * 2 + 1][j];
        unsigned mask = 0xffffffffu;
        float sel_v = 0.f; int sel_c = 0;
        for (int it = 0; it < TOPK; ++it) {
            float bv = NEG_BIG; int bj = 0;
#pragma unroll
            for (int j = 0; j < TOPK; ++j) {
                float cv = vl + tr[j];
                if (((mask >> j) & 1u) && cv > bv) { bv = cv; bj = j; }
            }
            int bc = (lane << 5) | bj;                 // flat position i*32+j (jax order)
#pragma unroll
            for (int off = 16; off > 0; off >>= 1) {
                float ov = __shfl_xor(bv, off, 32);
                int   oc = __shfl_xor(bc, off, 32);
                if (ov > bv || (ov == bv && oc < bc)) { bv = ov; bc = oc; }
            }
            if ((bc >> 5) == lane) mask &= ~(1u << (bc & 31));
            if (lane == it) { sel_v = bv; sel_c = bc; }  // lane k holds k-th largest
        }
        // softmax across the 32 selected (sorted desc -> lane 0 is the max)
        const float m = __shfl(sel_v, 0, 32);
        const float e = __expf(sel_v - m);
        float sum = e;
#pragma unroll
        for (int off = 16; off > 0; off >>= 1) sum += __shfl_xor(sum, off, 32);
        const int il = s_topi[h * 2][sel_c >> 5];
        const int ir = s_topi[h * 2 + 1][sel_c & 31];
        s_w [h][lane] = e / sum;
        s_vi[h][lane] = il * NSUB + ir;
    }
    __syncthreads();

    // ---- stage 3: weighted gather of value_table rows (float4 coalesced) ----
    const float4* vt = (const float4*)vtab;
    float4 acc = make_float4(0.f, 0.f, 0.f, 0.f);
    for (int h = 0; h < NHEAD; ++h) {
#pragma unroll 4
        for (int k = 0; k < TOPK; ++k) {
            const float w  = s_w[h][k];
            const int  row = s_vi[h][k];
            float4 v = vt[row * (DIM_IN / 4) + tid];   // 256 thr x 16B = full 4KB row
            acc.x += w * v.x; acc.y += w * v.y;
            acc.z += w * v.z; acc.w += w * v.w;
        }
    }
    ((float4*)out)[token * (DIM_IN / 4) + tid] = acc;
}

// ---------------------------------------------------------------------------
extern "C" void kernel_launch(void* const* d_in, const int* in_sizes, int n_in,
                              void* d_out, int out_size, void* d_ws, size_t ws_size,
                              hipStream_t stream) {
    const float* x     = (const float*)d_in[0];
    const float* Wq    = (const float*)d_in[1];
    const float* gamma = (const float*)d_in[2];
    const float* beta  = (const float*)d_in[3];
    const float* keyl  = (const float*)d_in[4];
    const float* keyr  = (const float*)d_in[5];
    const float* vtab  = (const float*)d_in[6];
    float* out = (float*)d_out;

    // workspace layout (~16.3 MB)
    float* q      = (float*)d_ws;                     // 1024 x 2048
    float* scale  = q + (size_t)NTOK * DIM_QF;        // 2048
    float* shift  = scale + DIM_QF;                   // 2048
    float* scores = shift + DIM_QF;                   // 1024 x 2048
    float* psum   = scores + (size_t)NTOK * DIM_QF;   // 16 x 2048
    float* psum2  = psum + (size_t)ROWCHUNK * DIM_QF; // 16 x 2048

    PKM_qproj      <<<dim3(NTOK / 16, DIM_QF / 32), 32, 0, stream>>>(x, Wq, q);
    PKM_bn_partial <<<dim3(DIM_QF / 256, ROWCHUNK), 256, 0, stream>>>(q, psum, psum2);
    PKM_bn_final   <<<DIM_QF / 256, 256, 0, stream>>>(psum, psum2, gamma, beta, scale, shift);
    PKM_bn_apply   <<<(NTOK * DIM_QF / 4) / 256, 256, 0, stream>>>(q, scale, shift);
    PKM_scores     <<<dim3(NTOK / 16, NSUB / 32, NHEAD * 2), 32, 0, stream>>>(q, keyl, keyr, scores);
    PKM_topk_gather<<<NTOK, 256, 0, stream>>>(scores, vtab, out);
}